// CURLNet_89472758710833
// MI455X (gfx1250) — compile-verified
//
#include <hip/hip_runtime.h>
#include <math.h>

// ---------------------------------------------------------------------------
// CURL-style image pipeline, fully fused.
//   out = hsv2rgb(adjustHSV(rgb2hsv(adjust3(lab2rgb(adjust3(img,L)),R)),S))
//   plus scalar ssd (parameter-only) at out[3*N].
// Each apply_curve reduces to scale = A + B*x (affine), so precompute 10
// (A,B) pairs once, then one streaming per-pixel kernel.
// Memory-bound: ~100 MB single-use traffic -> non-temporal B128 loads/stores
// + global_prefetch_b8 for the next grid-stride chunk.
// ---------------------------------------------------------------------------

typedef float v4f __attribute__((ext_vector_type(4)));  // native vector: OK for NT builtins

#define CLIP01(x) fminf(fmaxf((x), 0.0f), 1.0f)

__device__ __forceinline__ float nan0f(float x) { return (x == x) ? x : 0.0f; }
__device__ __forceinline__ float clip60(float x) { return fminf(fmaxf(x, 0.0f), 60.0f); }
__device__ __forceinline__ float pow_inv24(float x) {
    // x^(1/2.4) for x >= 1e-4: single v_log_f32 + v_exp_f32
    return __builtin_amdgcn_exp2f(0.4166666666666667f * __builtin_amdgcn_logf(x));
}

// --- kernel 1: curve coefficients + ssd (10 curves, one lane each) ----------
__global__ void curl_coeffs_kernel(const float* __restrict__ L,
                                   const float* __restrict__ R,
                                   const float* __restrict__ S,
                                   float* __restrict__ coeffs,   // A[10], B[10]
                                   float* __restrict__ ssd_out) {
    __shared__ float ssd_sh[10];
    const int lane = threadIdx.x;
    if (lane < 10) {
        const float* P;
        if (lane < 3)      P = L + lane * 16;
        else if (lane < 6) P = R + (lane - 3) * 16;
        else               P = S + (lane - 6) * 16;
        float C[16];
#pragma unroll
        for (int k = 0; k < 16; ++k) C[k] = __expf(P[k]);
        float slope[15];
#pragma unroll
        for (int j = 0; j < 15; ++j) slope[j] = C[j + 1] - C[j];
        float ssd = 0.0f;
#pragma unroll
        for (int j = 0; j < 14; ++j) { float d = slope[j + 1] - slope[j]; ssd += d * d; }
        float sum_s = 0.0f, dot = 0.0f;
#pragma unroll
        for (int j = 0; j < 14; ++j) { sum_s += slope[j]; dot += slope[j] * (float)j; }
        coeffs[lane]      = C[0] - dot;     // A
        coeffs[10 + lane] = 15.0f * sum_s;  // B = (K-1) * sum(s)
        ssd_sh[lane] = ssd;
    }
    __syncthreads();
    if (lane == 0) {
        float t = 0.0f;
#pragma unroll
        for (int i = 0; i < 10; ++i) t += ssd_sh[i];
        *ssd_out = t;
    }
}

// --- per-pixel fused pipeline ----------------------------------------------
__device__ __forceinline__ void adjust3(float& v0, float& v1, float& v2,
                                        const float* A, const float* B, int base) {
    // curve 0 reads RAW v0; later channels are clipped by the preceding
    // curve's full-tensor clip before being read (exact reference ordering).
    float s0 = A[base + 0] + B[base + 0] * v0; v0 = CLIP01(v0 * s0);
    float t1 = CLIP01(v1);
    float s1 = A[base + 1] + B[base + 1] * t1; v1 = CLIP01(t1 * s1);
    float t2 = CLIP01(v2);
    float s2 = A[base + 2] + B[base + 2] * t2; v2 = CLIP01(t2 * s2);
    v0 = nan0f(v0); v1 = nan0f(v1); v2 = nan0f(v2);
}

__device__ __forceinline__ void curl_pixel(float r, float g, float b,
                                           const float* A, const float* B,
                                           float& ro, float& go, float& bo) {
    // stage 1: adjust_3 with L-curves (0..2)
    adjust3(r, g, b, A, B, 0);

    // stage 2: lab_to_rgb   (r=L, g=a, b=b)
    {
        float Lv = r * 100.0f;
        float av = (g * 2.0f - 1.0f) * 110.0f;
        float bv = (b * 2.0f - 1.0f) * 110.0f;
        float fy = (Lv + 16.0f) * (1.0f / 116.0f);
        float fx = fy + av * (1.0f / 500.0f);
        float fz = fy - bv * (1.0f / 200.0f);
        const float eps  = 6.0f / 29.0f;
        const float k3e2 = 3.0f * eps * eps;
        float cx = fmaxf(fx, 1e-4f), cy = fmaxf(fy, 1e-4f), cz = fmaxf(fz, 1e-4f);
        float X = (fx <= eps) ? k3e2 * (fx - 4.0f / 29.0f) : cx * cx * cx;
        float Y = (fy <= eps) ? k3e2 * (fy - 4.0f / 29.0f) : cy * cy * cy;
        float Z = (fz <= eps) ? k3e2 * (fz - 4.0f / 29.0f) : cz * cz * cz;
        X *= 0.950456f; Z *= 1.088754f;
        float rr = X *  3.2404542f + Y * -1.5371385f + Z * -0.4985314f;
        float gg = X * -0.969266f  + Y *  1.8760108f + Z *  0.041556f;
        float bb = X *  0.0556434f + Y * -0.2040259f + Z *  1.0572252f;
        rr = (rr <= 0.0031308f) ? rr * 12.92f : 1.055f * pow_inv24(fmaxf(rr, 1e-4f)) - 0.055f;
        gg = (gg <= 0.0031308f) ? gg * 12.92f : 1.055f * pow_inv24(fmaxf(gg, 1e-4f)) - 0.055f;
        bb = (bb <= 0.0031308f) ? bb * 12.92f : 1.055f * pow_inv24(fmaxf(bb, 1e-4f)) - 0.055f;
        r = nan0f(rr); g = nan0f(gg); b = nan0f(bb);
    }

    // stage 3: adjust_3 with R-curves (3..5)
    adjust3(r, g, b, A, B, 3);

    // stage 4: rgb_to_hsv
    {
        float rc = fminf(fmaxf(r, 1e-9f), 1.0f);
        float gc = fminf(fmaxf(g, 1e-9f), 1.0f);
        float bc = fminf(fmaxf(b, 1e-9f), 1.0f);
        float mx = fmaxf(rc, fmaxf(gc, bc));
        float mn = fminf(rc, fminf(gc, bc));
        float df = mx - mn + 1e-10f;
        float inv = 60.0f / df;
        float h;
        if (mn == mx)       h = 0.0f;
        else if (rc == mx) { float t = (gc - bc) * inv; h = (t < 0.0f) ? t + 360.0f : t; }
        else if (gc == mx)  h = (bc - rc) * inv + 120.0f;   // in [60,180], no wrap
        else                h = (rc - gc) * inv + 240.0f;   // in [180,300], no wrap
        r = nan0f(h * (1.0f / 360.0f));
        g = nan0f(df / mx);
        b = nan0f(mx);
    }

    // stage 5: adjust_hsv, curves 6..9 with (cin,cout) = (0,0),(0,1),(1,1),(2,2)
    {
        float h = r, s = g, v = b;
        float sc = A[6] + B[6] * h; h = CLIP01(h * sc); s = CLIP01(s); v = CLIP01(v);
        sc = A[7] + B[7] * h; s = CLIP01(s * sc);
        sc = A[8] + B[8] * s; s = CLIP01(s * sc);
        sc = A[9] + B[9] * v; v = CLIP01(v * sc);
        r = nan0f(h); g = nan0f(s); b = nan0f(v);
    }

    // stage 6: hsv_to_rgb
    {
        float h = CLIP01(r), s = CLIP01(g), v = CLIP01(b);
        float c  = v * s * (1.0f / 60.0f);
        float hd = h * 360.0f;
        float rr = v - clip60(hd -  60.0f) * c + clip60(hd - 240.0f) * c;
        float gg = v * (1.0f - s) + clip60(hd) * c - clip60(hd - 180.0f) * c;
        float bb = v * (1.0f - s) + clip60(hd - 120.0f) * c - clip60(hd - 300.0f) * c;
        ro = CLIP01(nan0f(rr));
        go = CLIP01(nan0f(gg));
        bo = CLIP01(nan0f(bb));
    }
}

// --- kernel 2: streaming fused map (v4f vectorized, grid-stride) -----------
__global__ void __launch_bounds__(256)
curl_main_kernel(const float* __restrict__ img,
                 const float* __restrict__ coeffs,
                 float* __restrict__ out, int N /*pixels*/, int Nv /*v4f chunks*/) {
    __shared__ float cA[10], cB[10];
    const int tid = threadIdx.x;
    if (tid < 10) { cA[tid] = coeffs[tid]; cB[tid] = coeffs[10 + tid]; }
    __syncthreads();
    float A[10], B[10];
#pragma unroll
    for (int i = 0; i < 10; ++i) { A[i] = cA[i]; B[i] = cB[i]; }

    const v4f* r4 = (const v4f*)(img);
    const v4f* g4 = (const v4f*)(img + N);
    const v4f* b4 = (const v4f*)(img + 2 * (size_t)N);
    v4f* o0 = (v4f*)(out);
    v4f* o1 = (v4f*)(out + N);
    v4f* o2 = (v4f*)(out + 2 * (size_t)N);

    const int stride = gridDim.x * blockDim.x;
    for (int i = blockIdx.x * blockDim.x + tid; i < Nv; i += stride) {
        const int nx = i + stride;
        if (nx < Nv) {                       // prefetch next grid-stride chunk
            __builtin_prefetch(&r4[nx], 0, 3);   // -> global_prefetch_b8
            __builtin_prefetch(&g4[nx], 0, 3);
            __builtin_prefetch(&b4[nx], 0, 3);
        }
        // single-use streaming data: non-temporal (TH=NT) B128 accesses
        v4f rv = __builtin_nontemporal_load(&r4[i]);
        v4f gv = __builtin_nontemporal_load(&g4[i]);
        v4f bv = __builtin_nontemporal_load(&b4[i]);
        v4f ro, go, bo;
#pragma unroll
        for (int k = 0; k < 4; ++k) {
            float x, y, z;
            curl_pixel(rv[k], gv[k], bv[k], A, B, x, y, z);
            ro[k] = x; go[k] = y; bo[k] = z;
        }
        __builtin_nontemporal_store(ro, &o0[i]);
        __builtin_nontemporal_store(go, &o1[i]);
        __builtin_nontemporal_store(bo, &o2[i]);
    }
    // scalar tail (N not divisible by 4; no-op for 2048x2048)
    for (int p = (Nv << 2) + blockIdx.x * blockDim.x + tid; p < N; p += stride) {
        float r = img[p], g = img[N + p], b = img[2 * (size_t)N + p];
        float ro, go, bo;
        curl_pixel(r, g, b, A, B, ro, go, bo);
        out[p] = ro; out[N + p] = go; out[2 * (size_t)N + p] = bo;
    }
}

extern "C" void kernel_launch(void* const* d_in, const int* in_sizes, int n_in,
                              void* d_out, int out_size, void* d_ws, size_t ws_size,
                              hipStream_t stream) {
    const float* img = (const float*)d_in[0];
    const float* L   = (const float*)d_in[1];
    const float* R   = (const float*)d_in[2];
    const float* S   = (const float*)d_in[3];
    float* out = (float*)d_out;
    float* coeffs = (float*)d_ws;              // 20 floats of scratch

    const int N  = in_sizes[0] / 3;            // pixels per channel
    const int Nv = ((N & 3) == 0) ? (N >> 2) : 0;

    // 1) curve coefficients + ssd -> out[3N]
    curl_coeffs_kernel<<<1, 32, 0, stream>>>(L, R, S, coeffs, out + (size_t)3 * N);

    // 2) fused streaming map
    const int threads = 256;
    int work = (Nv > 0) ? Nv : N;
    int blocks = (work + threads - 1) / threads;
    if (blocks > 2048) blocks = 2048;          // >=2 grid-stride iters so prefetch helps
    if (blocks < 1) blocks = 1;
    curl_main_kernel<<<blocks, threads, 0, stream>>>(img, coeffs, out, N, Nv);
}